// AdditiveAttention_17446157156499
// MI455X (gfx1250) — compile-verified
//
#include <hip/hip_runtime.h>
#include <hip/hip_bf16.h>

// Problem constants (match reference)
#define BATCH 4
#define QLEN  512
#define KN    1024
#define DDIM  256    // QS == KS
#define HDIM  128    // H
#define VDIM  256    // VD
#define NEGV  (-1e9f)

typedef __attribute__((ext_vector_type(2))) float v2f;
typedef __attribute__((ext_vector_type(8))) float v8f;

__device__ __forceinline__ float fast_tanh(float x) {
#if __has_builtin(__builtin_amdgcn_tanhf)
  return __builtin_amdgcn_tanhf(x);   // v_tanh_f32 (CDNA5 transcendental)
#else
  return tanhf(x);
#endif
}

// ---------------------------------------------------------------------------
// Kernel 1/2: out(M,128) = A(M,256) x W(256,128), f32 WMMA 16x16x4.
// Block = 256 threads = 8 waves. Block owns one 16-row M tile; wave w owns
// N-tile [w*16, w*16+16). A tile staged in LDS (padded stride 260 -> float4
// aligned rows, conflict-free column fragment reads).
// ---------------------------------------------------------------------------
__global__ void __launch_bounds__(256)
proj_gemm_kernel(const float* __restrict__ A,
                 const float* __restrict__ W,
                 float* __restrict__ out)
{
  __shared__ __align__(16) float Ash[16][260];   // 16 x 256 padded

  const int tid  = threadIdx.x;
  const int wave = tid >> 5;
  const int lane = tid & 31;
  const int m0   = blockIdx.x * 16;

  // Cooperative, coalesced A tile load: 16*64 float4
  for (int i = tid; i < 16 * 64; i += 256) {
    const int r  = i >> 6;
    const int c4 = i & 63;
    const float4 v = ((const float4*)(A + (size_t)(m0 + r) * DDIM))[c4];
    *((float4*)&Ash[r][c4 * 4]) = v;
  }
  __syncthreads();

  const int n0   = wave * 16;          // 8 waves cover N = 128
  const int row  = lane & 15;
  const int kofs = (lane >> 4) << 1;   // lanes 0-15 -> K+0/1, lanes 16-31 -> K+2/3

  v8f acc = {};
  for (int k = 0; k < DDIM; k += 4) {
    v2f a, b;
    a.x = Ash[row][k + kofs];
    a.y = Ash[row][k + kofs + 1];
    const float* wp = W + (size_t)(k + kofs) * HDIM + n0 + row;
    b.x = wp[0];
    b.y = wp[HDIM];
    acc = __builtin_amdgcn_wmma_f32_16x16x4_f32(false, a, false, b,
                                                (short)0, acc, false, false);
  }

  // D layout: lanes 0-15 -> rows m0..m0+7, lanes 16-31 -> rows m0+8..m0+15
  const int mbase = m0 + ((lane >> 4) * 8);
  float* op = out + (size_t)mbase * HDIM + n0 + row;
#pragma unroll
  for (int v = 0; v < 8; v++) op[(size_t)v * HDIM] = acc[v];
}

// ---------------------------------------------------------------------------
// Kernel 3: per (batch, 16-query tile):
//   pass1: scores[k][qi] = sum_h w[h]*tanh(qh[qi][h]+kh[k][h])  (v_tanh_f32)
//   pass2: row softmax over K=1024 (wave32 shfl reductions), keep 1/rowsum
//   pass3: out(16x256) = P(16x1024) x V(1024x256) via f32 WMMA, scaled store
// ---------------------------------------------------------------------------
__global__ void __launch_bounds__(256)
attn_kernel(const float* __restrict__ qh,       // (B*Q, 128)
            const float* __restrict__ kh,       // (B*K, 128)
            const float* __restrict__ V,        // (B, K, 256)
            const int*   __restrict__ valid_lens,
            const float* __restrict__ w_v,      // (128)
            float* __restrict__ out)            // (B, Q, 256)
{
  __shared__ __align__(16) float scT[KN][16];   // 64 KB, transposed P
  __shared__ __align__(16) float qsh[16][HDIM]; // 8 KB
  __shared__ float wsh[HDIM];
  __shared__ float rowscale[16];

  const int b   = blockIdx.y;
  const int qt  = blockIdx.x;
  const int tid = threadIdx.x;
  const int wave = tid >> 5;
  const int lane = tid & 31;
  const int vlen = valid_lens[b];

  // Stage query tile + w_v
  for (int i = tid; i < 16 * (HDIM / 4); i += 256) {
    const int r = i >> 5;
    const int c = i & 31;
    ((float4*)qsh[r])[c] =
        ((const float4*)(qh + (size_t)(b * QLEN + qt * 16 + r) * HDIM))[c];
  }
  if (tid < HDIM) wsh[tid] = w_v[tid];
  __syncthreads();

  // ---- Pass 1: additive scores with hardware tanh ----
  const float* khb = kh + (size_t)b * KN * HDIM;
  for (int kk = 0; kk < 4; kk++) {
    const int k = tid + kk * 256;
    const float* kp = khb + (size_t)k * HDIM;
    float s[16];
#pragma unroll
    for (int qi = 0; qi < 16; qi++) s[qi] = 0.f;

    for (int hc = 0; hc < HDIM; hc += 16) {
      const float4 kv0 = *(const float4*)(kp + hc);
      const float4 kv1 = *(const float4*)(kp + hc + 4);
      const float4 kv2 = *(const float4*)(kp + hc + 8);
      const float4 kv3 = *(const float4*)(kp + hc + 12);
      const float4 w0 = *(const float4*)(wsh + hc);
      const float4 w1 = *(const float4*)(wsh + hc + 4);
      const float4 w2 = *(const float4*)(wsh + hc + 8);
      const float4 w3 = *(const float4*)(wsh + hc + 12);
#pragma unroll
      for (int qi = 0; qi < 16; qi++) {
        const float4 q0 = *(const float4*)(&qsh[qi][hc]);
        const float4 q1 = *(const float4*)(&qsh[qi][hc + 4]);
        const float4 q2 = *(const float4*)(&qsh[qi][hc + 8]);
        const float4 q3 = *(const float4*)(&qsh[qi][hc + 12]);
        float acc = s[qi];
        acc += w0.x * fast_tanh(q0.x + kv0.x);
        acc += w0.y * fast_tanh(q0.y + kv0.y);
        acc += w0.z * fast_tanh(q0.z + kv0.z);
        acc += w0.w * fast_tanh(q0.w + kv0.w);
        acc += w1.x * fast_tanh(q1.x + kv1.x);
        acc += w1.y * fast_tanh(q1.y + kv1.y);
        acc += w1.z * fast_tanh(q1.z + kv1.z);
        acc += w1.w * fast_tanh(q1.w + kv1.w);
        acc += w2.x * fast_tanh(q2.x + kv2.x);
        acc += w2.y * fast_tanh(q2.y + kv2.y);
        acc += w2.z * fast_tanh(q2.z + kv2.z);
        acc += w2.w * fast_tanh(q2.w + kv2.w);
        acc += w3.x * fast_tanh(q3.x + kv3.x);
        acc += w3.y * fast_tanh(q3.y + kv3.y);
        acc += w3.z * fast_tanh(q3.z + kv3.z);
        acc += w3.w * fast_tanh(q3.w + kv3.w);
        s[qi] = acc;
      }
    }

    const bool valid = (k < vlen);
#pragma unroll
    for (int qi = 0; qi < 16; qi++) if (!valid) s[qi] = NEGV;
    float4* dst = (float4*)scT[k];
    dst[0] = make_float4(s[0],  s[1],  s[2],  s[3]);
    dst[1] = make_float4(s[4],  s[5],  s[6],  s[7]);
    dst[2] = make_float4(s[8],  s[9],  s[10], s[11]);
    dst[3] = make_float4(s[12], s[13], s[14], s[15]);
  }
  __syncthreads();

  // ---- Pass 2: softmax per query row (each wave: 2 rows) ----
  for (int r = 0; r < 2; r++) {
    const int qi = wave * 2 + r;
    float m = -3.402823466e38f;
    for (int k = lane; k < KN; k += 32) m = fmaxf(m, scT[k][qi]);
#pragma unroll
    for (int off = 16; off > 0; off >>= 1) m = fmaxf(m, __shfl_xor(m, off, 32));
    float sum = 0.f;
    for (int k = lane; k < KN; k += 32) {
      const float p = __expf(scT[k][qi] - m);
      scT[k][qi] = p;
      sum += p;
    }
#pragma unroll
    for (int off = 16; off > 0; off >>= 1) sum += __shfl_xor(sum, off, 32);
    if (lane == 0) rowscale[qi] = 1.0f / sum;
  }
  __syncthreads();

  // ---- Pass 3: out = P x V with f32 WMMA; wave owns 32 output columns ----
  const int n0   = wave * 32;
  const int row  = lane & 15;
  const int kofs = (lane >> 4) << 1;
  const float* Vb = V + (size_t)b * KN * VDIM;

  v8f acc0 = {};
  v8f acc1 = {};
  for (int k = 0; k < KN; k += 4) {
    v2f a;
    a.x = scT[k + kofs][row];          // conflict-free: 32 lanes -> 32 banks
    a.y = scT[k + kofs + 1][row];
    const float* vp = Vb + (size_t)(k + kofs) * VDIM + n0 + row;
    v2f b0, b1;
    b0.x = vp[0];
    b0.y = vp[VDIM];
    b1.x = vp[16];
    b1.y = vp[VDIM + 16];
    acc0 = __builtin_amdgcn_wmma_f32_16x16x4_f32(false, a, false, b0,
                                                 (short)0, acc0, false, false);
    acc1 = __builtin_amdgcn_wmma_f32_16x16x4_f32(false, a, false, b1,
                                                 (short)0, acc1, false, false);
  }

  const int mrow = (lane >> 4) * 8;
  float* op = out + ((size_t)b * QLEN + qt * 16 + mrow) * VDIM + n0 + row;
#pragma unroll
  for (int v = 0; v < 8; v++) {
    const float sc = rowscale[mrow + v];
    op[(size_t)v * VDIM]      = acc0[v] * sc;
    op[(size_t)v * VDIM + 16] = acc1[v] * sc;
  }
}

// ---------------------------------------------------------------------------
extern "C" void kernel_launch(void* const* d_in, const int* in_sizes, int n_in,
                              void* d_out, int out_size, void* d_ws, size_t ws_size,
                              hipStream_t stream) {
  const float* queries    = (const float*)d_in[0];  // (B,Q,256)
  const float* keys       = (const float*)d_in[1];  // (B,K,256)
  const float* values     = (const float*)d_in[2];  // (B,K,256)
  const int*   valid_lens = (const int*)  d_in[3];  // (B,)
  const float* W_q        = (const float*)d_in[4];  // (256,128)
  const float* W_k        = (const float*)d_in[5];  // (256,128)
  const float* w_v        = (const float*)d_in[6];  // (128,)
  float* out = (float*)d_out;                       // (B,Q,256)

  float* qh = (float*)d_ws;                                   // (B*Q,128) = 1 MB
  float* kh = qh + (size_t)BATCH * QLEN * HDIM;               // (B*K,128) = 2 MB

  proj_gemm_kernel<<<dim3((BATCH * QLEN) / 16), 256, 0, stream>>>(queries, W_q, qh);
  proj_gemm_kernel<<<dim3((BATCH * KN)   / 16), 256, 0, stream>>>(keys,    W_k, kh);
  attn_kernel<<<dim3(QLEN / 16, BATCH), 256, 0, stream>>>(qh, kh, values,
                                                          valid_lens, w_v, out);
}